// yololoss_53652731461925
// MI455X (gfx1250) — compile-verified
//
#include <hip/hip_runtime.h>
#include <hip/hip_bf16.h>

// ---------------- problem constants (match reference) ----------------
#define S_ 7
#define B_ 2
#define C_ 20
#define CELLF 30                    // C + 5*B floats per grid cell
#define NSAMP 16384
#define NCELLS (NSAMP * S_ * S_)    // 802816 cells
#define BLK 256                     // 8 wave32 per block
#define GRID (NCELLS / BLK)         // 3136 blocks (exact)
#define LAMBDA_COORD 5.0f
#define LAMBDA_NOOBJ 0.5f

typedef __attribute__((ext_vector_type(2))) float v2f;
typedef __attribute__((ext_vector_type(8))) float v8f;

__device__ __forceinline__ float sq(float x) { return x * x; }

// jnp.sign semantics: sign(0) == 0
__device__ __forceinline__ float sgnf(float x) {
    return (x > 0.f ? 1.f : 0.f) - (x < 0.f ? 1.f : 0.f);
}

__device__ __forceinline__ float iou4(const float* b1, const float* b2) {
    float b1x1 = b1[0] - b1[2] * 0.5f, b1y1 = b1[1] - b1[3] * 0.5f;
    float b1x2 = b1[0] + b1[2] * 0.5f, b1y2 = b1[1] + b1[3] * 0.5f;
    float b2x1 = b2[0] - b2[2] * 0.5f, b2y1 = b2[1] - b2[3] * 0.5f;
    float b2x2 = b2[0] + b2[2] * 0.5f, b2y2 = b2[1] + b2[3] * 0.5f;
    float x1 = fmaxf(b1x1, b2x1), y1 = fmaxf(b1y1, b2y1);
    float x2 = fminf(b1x2, b2x2), y2 = fminf(b1y2, b2y2);
    float inter = fmaxf(x2 - x1, 0.f) * fmaxf(y2 - y1, 0.f);
    float a1 = fabsf((b1x2 - b1x1) * (b1y2 - b1y1));
    float a2 = fabsf((b2x2 - b2x1) * (b2y2 - b2y1));
    return inter / (a1 + a2 - inter + 1e-6f);
}

// Cross-lane sum of one float per lane using V_WMMA_F32_16X16X4_F32:
// A(16x4) holds the 32 lane values (a.y = 0), B(4x16) = ones, so every
// column of D holds the sum split across lane n (rows 0-7) and lane n+16
// (rows 8-15). Result valid in lanes 0..15 (use lane 0).
__device__ __forceinline__ float wave_sum_wmma(float v) {
    v2f a; a.x = v;  a.y = 0.f;
    v2f b; b.x = 1.f; b.y = 1.f;
    v8f c = {};
    c = __builtin_amdgcn_wmma_f32_16x16x4_f32(false, a, false, b,
                                              (short)0, c, false, false);
    float s = c[0] + c[1] + c[2] + c[3] + c[4] + c[5] + c[6] + c[7];
    s += __shfl_down(s, 16, 32);
    return s;
}

__global__ __launch_bounds__(BLK) void yolo_loss_partial(
    const float* __restrict__ P, const float* __restrict__ T,
    float* __restrict__ part)
{
    const int cell = blockIdx.x * BLK + threadIdx.x;   // grid sized exactly

    // Cell stride is 120 bytes -> float2 (8B) aligned for every cell.
    const float2* pp = reinterpret_cast<const float2*>(P) + (size_t)cell * (CELLF / 2);
    const float2* tp = reinterpret_cast<const float2*>(T) + (size_t)cell * (CELLF / 2);

    float p[CELLF], t[CELLF];
#pragma unroll
    for (int i = 0; i < CELLF / 2; ++i) { float2 v = pp[i]; p[2*i] = v.x; p[2*i+1] = v.y; }
#pragma unroll
    for (int i = 0; i < CELLF / 2; ++i) { float2 v = tp[i]; t[2*i] = v.x; t[2*i+1] = v.y; }

    // ---- per-cell YOLO loss (exactly mirrors reference) ----
    const float iou1 = iou4(p + 21, t + 21);
    const float iou2 = iou4(p + 26, t + 21);
    const float best = (iou2 > iou1) ? 1.f : 0.f;
    const float omb  = 1.f - best;
    const float ex   = t[20];
    const float ne   = 1.f - ex;

    float bpx = ex * (best * p[26] + omb * p[21]);
    float bpy = ex * (best * p[27] + omb * p[22]);
    float bpw = ex * (best * p[28] + omb * p[23]);
    float bph = ex * (best * p[29] + omb * p[24]);

    const float sw = sgnf(bpw) * sqrtf(fabsf(bpw + 1e-6f));
    const float sh = sgnf(bph) * sqrtf(fabsf(bph + 1e-6f));

    float box_loss = sq(bpx - ex * t[21]) + sq(bpy - ex * t[22])
                   + sq(sw - sqrtf(ex * t[23])) + sq(sh - sqrtf(ex * t[24]));

    const float pconf    = best * p[25] + omb * p[20];
    const float obj_loss = sq(ex * pconf - ex * t[20]);

    const float noobj_loss = sq(ne * p[20] - ne * t[20])
                           + sq(ne * p[25] - ne * t[20]);

    float cls_loss = 0.f;
#pragma unroll
    for (int i = 0; i < C_; ++i) cls_loss += sq(ex * p[i] - ex * t[i]);

    const float loss = LAMBDA_COORD * box_loss + obj_loss
                     + LAMBDA_NOOBJ * noobj_loss + cls_loss;

    // ---- block reduction: WMMA wave-sum, then LDS across the 8 waves ----
    __shared__ float wsum[BLK / 32];
    const float s    = wave_sum_wmma(loss);
    const int   lane = threadIdx.x & 31;
    const int   wid  = threadIdx.x >> 5;
    if (lane == 0) wsum[wid] = s;
    __syncthreads();
    if (threadIdx.x == 0) {
        float bs = 0.f;
#pragma unroll
        for (int i = 0; i < BLK / 32; ++i) bs += wsum[i];
        part[blockIdx.x] = bs;
    }
}

// Single-wave finisher: chain C += A(64 partials) x ones over 3136 = 49*64
// block partials, entirely inside the matrix pipe, FP32 throughout.
__global__ __launch_bounds__(32) void yolo_loss_final(
    const float* __restrict__ part, float* __restrict__ out)
{
    const int lane = threadIdx.x;
    v2f b; b.x = 1.f; b.y = 1.f;
    v8f c = {};
    for (int base = 0; base < GRID; base += 64) {
        float2 v = reinterpret_cast<const float2*>(part)[base / 2 + lane];
        v2f a; a.x = v.x; a.y = v.y;
        c = __builtin_amdgcn_wmma_f32_16x16x4_f32(false, a, false, b,
                                                  (short)0, c, false, false);
    }
    float s = c[0] + c[1] + c[2] + c[3] + c[4] + c[5] + c[6] + c[7];
    s += __shfl_down(s, 16, 32);
    if (lane == 0) out[0] = s;
}

extern "C" void kernel_launch(void* const* d_in, const int* in_sizes, int n_in,
                              void* d_out, int out_size, void* d_ws, size_t ws_size,
                              hipStream_t stream) {
    const float* P = (const float*)d_in[0];   // predictions (N, 1470) f32
    const float* T = (const float*)d_in[1];   // target (N,7,7,30) f32, same flat layout
    float* part = (float*)d_ws;               // 3136 floats of scratch

    yolo_loss_partial<<<GRID, BLK, 0, stream>>>(P, T, part);
    yolo_loss_final<<<1, 32, 0, stream>>>(part, (float*)d_out);
}